// CompGCN_25898652794883
// MI455X (gfx1250) — compile-verified
//
#include <hip/hip_runtime.h>
#include <hip/hip_bf16.h>
#include <math.h>

#define N_NODES  100000
#define N_EDGES  1000000
#define EHALF    500000
#define H        128
#define N_REL    200
#define N_RELG   100
#define N_GRAPHS 256
#define N_CLS    2

typedef float v2f __attribute__((ext_vector_type(2)));
typedef float v8f __attribute__((ext_vector_type(8)));
typedef unsigned int v4u __attribute__((ext_vector_type(4)));
typedef int v4i __attribute__((ext_vector_type(4)));
typedef int v8i __attribute__((ext_vector_type(8)));

// -------- pure f32 WMMA: D(16x16,f32) = A(16x4,f32) x B(4x16,f32) + C --------
__device__ __forceinline__ v8f wmma_f32(v2f a, v2f b, v8f c) {
  return __builtin_amdgcn_wmma_f32_16x16x4_f32(false, a, false, b, (short)0, c,
                                               false, false);
}

// -------- stage one 128x128 f32 weight matrix into LDS via the Tensor Data Mover --------
// D# per CDNA5 ISA ch.8: group0 = {count|flags, lds_addr, ga_lo, ga_hi|type=2};
// group1 packs data_size=4B, tensor_dim0/1=128, tile_dim0/1=128, dim0_stride=128.
// This toolchain exposes the 6-arg builtin: (v4u g0, v8i g1, v4i, v4i, v8i, i32 cpol).
__device__ __forceinline__ void stage_weight(const float* __restrict__ gw,
                                             float* lds_buf, int wid) {
#if defined(__AMDGCN__) && __has_builtin(__builtin_amdgcn_tensor_load_to_lds) && \
    __has_builtin(__builtin_amdgcn_s_wait_tensorcnt)
  if (wid == 0) {                        // one DMA per block; TDM ignores EXEC
    unsigned long long ga = (unsigned long long)(uintptr_t)gw;
    unsigned int ldsoff = (unsigned int)(uintptr_t)lds_buf;  // generic addr low 32 = LDS offset
    v4u g0 = { 1u,                                            // count=1, user descriptor
               ldsoff,                                        // lds_addr
               (unsigned int)ga,                              // global_addr[31:0]
               ((unsigned int)(ga >> 32) & 0x01FFFFFFu) | (2u << 30) }; // ga[56:32], type=2
    v8i g1 = { (int)(2u << 16),       // workgroup_mask=0, data_size=2 (4 bytes)
               (int)(128u << 16),     // tensor_dim0 = 128 (bits 79:48 low half)
               (int)(128u << 16),     // tensor_dim1 = 128 (bits 111:80 low half)
               (int)(128u << 16),     // tile_dim0   = 128 (bits 127:112)
               128,                   // tile_dim1   = 128, tile_dim2 = 0
               128,                   // tensor_dim0_stride = 128 (low 32)
               0, 0 };
    v4i z4 = { 0, 0, 0, 0 };          // groups 2/3 unused (2D tensor)
    v8i z8 = { 0, 0, 0, 0, 0, 0, 0, 0 };
    __builtin_amdgcn_tensor_load_to_lds(g0, g1, z4, z4, z8, 0);
    __builtin_amdgcn_s_wait_tensorcnt(0);
  }
#else
  for (int i = threadIdx.x; i < (H * H) / 4; i += 256)
    ((float4*)lds_buf)[i] = ((const float4*)gw)[i];
#endif
}

// ---------------- degree / norm (layer-invariant) ----------------
__global__ void deg_kernel(const int* __restrict__ ei, float* __restrict__ deg) {
  int e = blockIdx.x * blockDim.x + threadIdx.x;
  if (e >= N_EDGES) return;
  float* d = deg + (e < EHALF ? 0 : N_NODES);
  atomicAdd(d + ei[e], 1.0f);
}

__global__ void norm_kernel(const int* __restrict__ ei, const float* __restrict__ deg,
                            float* __restrict__ norm) {
  int e = blockIdx.x * blockDim.x + threadIdx.x;
  if (e >= N_EDGES) return;
  int src = ei[e];
  int dst = ei[N_EDGES + e];
  const float* d = deg + (e < EHALF ? 0 : N_NODES);
  float ds = d[src], dd = d[dst];
  float is = ds > 0.f ? rsqrtf(ds) : 0.f;
  float id = dd > 0.f ? rsqrtf(dd) : 0.f;
  norm[e] = is * id;
}

// ---------------- rel buffers ----------------
__global__ void build_rel1(const float* __restrict__ rg, const float* __restrict__ lr,
                           float* __restrict__ rel) {
  int t = blockIdx.x * blockDim.x + threadIdx.x;
  if (t >= 201 * H) return;
  if (t < N_RELG * H)          rel[t] = rg[t];
  else if (t < 2 * N_RELG * H) rel[t] = -rg[t - N_RELG * H];
  else                         rel[t] = lr[t - 2 * N_RELG * H];
}

__global__ void set_loop_row(const float* __restrict__ lr, float* __restrict__ rel) {
  int t = threadIdx.x;
  if (t < H) rel[200 * H + t] = lr[t];
}

// ---------------- edge gather-compose-scatter (one wave per edge) ----------------
__global__ void __launch_bounds__(256)
edge_agg_kernel(const float* __restrict__ x, const float* __restrict__ rel,
                const int* __restrict__ ei, const int* __restrict__ et,
                const float* __restrict__ norm,
                float* __restrict__ agg_in, float* __restrict__ agg_out) {
  int wid  = threadIdx.x >> 5;
  int lane = threadIdx.x & 31;
  int e = blockIdx.x * (blockDim.x >> 5) + wid;
  if (e >= N_EDGES) return;
  int src = ei[e];
  int dst = ei[N_EDGES + e];
  int t   = et[e];
  float nrm = norm[e];
  float* agg = (e < EHALF) ? agg_in : agg_out;
  int h = lane * 4;
  const float4 xv = *(const float4*)(x + (size_t)src * H + h);
  const float4 rv = *(const float4*)(rel + (size_t)t * H + h);
  float* dp = agg + (size_t)dst * H + h;
  atomicAdd(dp + 0, (xv.x - rv.x) * nrm);
  atomicAdd(dp + 1, (xv.y - rv.y) * nrm);
  atomicAdd(dp + 2, (xv.z - rv.z) * nrm);
  atomicAdd(dp + 3, (xv.w - rv.w) * nrm);
}

// ---- fused layer GEMM: wave = 16 rows x 128 cols, K = 3*128, B staged in LDS by TDM ----
__global__ void __launch_bounds__(256)
fused_layer_kernel(const float* __restrict__ agg_in, const float* __restrict__ agg_out,
                   const float* __restrict__ xin,
                   const float* __restrict__ w_in, const float* __restrict__ w_out,
                   const float* __restrict__ w_loop,
                   const float* __restrict__ loop_rel, const float* __restrict__ bias,
                   float* __restrict__ xout, int do_relu) {
  __shared__ float w_lds[H * H];           // 64 KB: current weight matrix

  int wid  = threadIdx.x >> 5;
  int lane = threadIdx.x & 31;
  int tile = blockIdx.x * 8 + wid;
  bool valid = (tile < N_NODES / 16);
  if (!valid) tile = 0;                    // keep tail waves in step for barriers
  int mrow  = lane & 15;                   // A-row / B,D-column sub-index
  int khalf = lane >> 4;                   // which K-pair this lane holds
  int arow  = tile * 16 + mrow;

  v8f acc[8];
#pragma unroll
  for (int n = 0; n < 8; ++n) { v8f z = {}; acc[n] = z; }

  const float* Aseg[3] = { agg_in, agg_out, xin };
  const float* Wseg[3] = { w_in,   w_out,   w_loop };
#pragma unroll
  for (int s = 0; s < 3; ++s) {
    stage_weight(Wseg[s], w_lds, wid);     // TDM DMA + s_wait_tensorcnt (wave 0)
    __syncthreads();
    const float* __restrict__ Ap = Aseg[s] + (size_t)arow * H;
#pragma unroll 4
    for (int k0 = 0; k0 < H; k0 += 4) {
      int kk = k0 + khalf * 2;
      v2f a = *(const v2f*)(Ap + kk);                 // A[row][kk], A[row][kk+1]
      if (s == 2) { a.x -= loop_rel[kk]; a.y -= loop_rel[kk + 1]; }  // (x - loop_rel)
#pragma unroll
      for (int n = 0; n < 8; ++n) {
        int col = n * 16 + mrow;
        v2f b;
        b.x = w_lds[kk * H + col];                    // B[kk][col]   (ds_load)
        b.y = w_lds[(kk + 1) * H + col];              // B[kk+1][col] (ds_load)
        acc[n] = wmma_f32(a, b, acc[n]);
      }
    }
    __syncthreads();                       // all waves done before next restage
  }

  const float inv3 = 1.0f / 3.0f;
#pragma unroll
  for (int n = 0; n < 8; ++n) {
    int col = n * 16 + mrow;
    float bv = bias[col];
#pragma unroll
    for (int v = 0; v < 8; ++v) {
      int row = tile * 16 + khalf * 8 + v;            // D layout: M = v + 8*(lane>=16)
      float t = tanhf(acc[n][v] * inv3 + bv);
      if (do_relu) t = fmaxf(t, 0.0f);
      if (valid) xout[(size_t)row * H + col] = t;
    }
  }
}

// ---------------- rel update GEMM: rel_next(200x128) = rel(201x128) @ w_rel ----------------
__global__ void rel_gemm_kernel(const float* __restrict__ rel,
                                const float* __restrict__ w_rel,
                                float* __restrict__ rel_next) {
  int lane  = threadIdx.x & 31;
  int tile  = blockIdx.x;                 // 13 tiles cover 208 rows (201 valid)
  int mrow  = lane & 15;
  int khalf = lane >> 4;
  int arow  = tile * 16 + mrow;

  v8f acc[8];
#pragma unroll
  for (int n = 0; n < 8; ++n) { v8f z = {}; acc[n] = z; }

#pragma unroll 4
  for (int k0 = 0; k0 < H; k0 += 4) {
    int kk = k0 + khalf * 2;
    v2f a;
    if (arow < 201) a = *(const v2f*)(rel + (size_t)arow * H + kk);
    else            { a.x = 0.f; a.y = 0.f; }
#pragma unroll
    for (int n = 0; n < 8; ++n) {
      int col = n * 16 + mrow;
      v2f b;
      b.x = w_rel[(size_t)kk * H + col];
      b.y = w_rel[(size_t)(kk + 1) * H + col];
      acc[n] = wmma_f32(a, b, acc[n]);
    }
  }
#pragma unroll
  for (int n = 0; n < 8; ++n) {
    int col = n * 16 + mrow;
#pragma unroll
    for (int v = 0; v < 8; ++v) {
      int row = tile * 16 + khalf * 8 + v;
      if (row < N_REL) rel_next[(size_t)row * H + col] = acc[n][v];
    }
  }
}

// ---------------- mean-pool + head ----------------
__global__ void __launch_bounds__(256)
pool_kernel(const float* __restrict__ x, const int* __restrict__ batch,
            float* __restrict__ pool, float* __restrict__ cnt) {
  int wid  = threadIdx.x >> 5;
  int lane = threadIdx.x & 31;
  int node = blockIdx.x * (blockDim.x >> 5) + wid;
  if (node >= N_NODES) return;
  int g = batch[node];
  int h = lane * 4;
  const float4 xv = *(const float4*)(x + (size_t)node * H + h);
  float* p = pool + (size_t)g * H + h;
  atomicAdd(p + 0, xv.x); atomicAdd(p + 1, xv.y);
  atomicAdd(p + 2, xv.z); atomicAdd(p + 3, xv.w);
  if (lane == 0) atomicAdd(cnt + g, 1.0f);
}

__global__ void head_kernel(const float* __restrict__ pool, const float* __restrict__ cnt,
                            const float* __restrict__ rel_emb_table,
                            const int* __restrict__ rel_labels,
                            const float* __restrict__ lin_w, const float* __restrict__ lin_b,
                            float* __restrict__ out) {
  int g = blockIdx.x * blockDim.x + threadIdx.x;
  if (g >= N_GRAPHS) return;
  float inv = 1.0f / fmaxf(cnt[g], 1.0f);
  const float* re = rel_emb_table + (size_t)rel_labels[g] * H;
  float a0 = lin_b[0], a1 = lin_b[1];
  for (int h = 0; h < H; ++h) {
    float m = pool[(size_t)g * H + h] * inv;
    a0 += m * lin_w[h * N_CLS + 0];
    a1 += m * lin_w[h * N_CLS + 1];
    float r = re[h];
    a0 += r * lin_w[(H + h) * N_CLS + 0];
    a1 += r * lin_w[(H + h) * N_CLS + 1];
  }
  out[g * N_CLS + 0] = a0;
  out[g * N_CLS + 1] = a1;
}

extern "C" void kernel_launch(void* const* d_in, const int* in_sizes, int n_in,
                              void* d_out, int out_size, void* d_ws, size_t ws_size,
                              hipStream_t stream) {
  (void)in_sizes; (void)n_in; (void)out_size; (void)ws_size;
  const float* x0            = (const float*)d_in[0];
  const int*   ei            = (const int*)d_in[1];
  const int*   et            = (const int*)d_in[2];
  const int*   batch         = (const int*)d_in[3];
  const int*   rel_labels    = (const int*)d_in[4];
  const float* rel_emb_table = (const float*)d_in[6];
  const float* rel_graph_emb = (const float*)d_in[7];
  const float* w_in[3]   = {(const float*)d_in[8],  (const float*)d_in[14], (const float*)d_in[20]};
  const float* w_out[3]  = {(const float*)d_in[9],  (const float*)d_in[15], (const float*)d_in[21]};
  const float* w_loop[3] = {(const float*)d_in[10], (const float*)d_in[16], (const float*)d_in[22]};
  const float* w_rel[3]  = {(const float*)d_in[11], (const float*)d_in[17], (const float*)d_in[23]};
  const float* loop_rel[3] = {(const float*)d_in[12], (const float*)d_in[18], (const float*)d_in[24]};
  const float* bias[3]   = {(const float*)d_in[13], (const float*)d_in[19], (const float*)d_in[25]};
  const float* lin_w = (const float*)d_in[26];
  const float* lin_b = (const float*)d_in[27];
  float* out = (float*)d_out;

  float* ws = (float*)d_ws;
  size_t off = 0;
  float* norm   = ws + off; off += N_EDGES;
  float* deg    = ws + off; off += 2 * (size_t)N_NODES;
  float* aggin  = ws + off; off += (size_t)N_NODES * H;   // aggin+aggout contiguous
  float* aggout = ws + off; off += (size_t)N_NODES * H;
  float* x1     = ws + off; off += (size_t)N_NODES * H;
  float* x2     = ws + off; off += (size_t)N_NODES * H;
  float* relA   = ws + off; off += 201 * H;
  float* relB   = ws + off; off += 201 * H;
  float* pool   = ws + off; off += (size_t)N_GRAPHS * H;  // pool+cnt contiguous
  float* cnt    = ws + off; off += N_GRAPHS;

  // degrees + symmetric norms (depend only on edge_index)
  (void)hipMemsetAsync(deg, 0, 2 * (size_t)N_NODES * sizeof(float), stream);
  deg_kernel<<<(N_EDGES + 255) / 256, 256, 0, stream>>>(ei, deg);
  norm_kernel<<<(N_EDGES + 255) / 256, 256, 0, stream>>>(ei, deg, norm);

  // layer-1 relations: [rel_graph_emb; -rel_graph_emb; loop_rel1]
  build_rel1<<<(201 * H + 255) / 256, 256, 0, stream>>>(rel_graph_emb, loop_rel[0], relA);

  const float* xin = x0;
  float* xbuf[2] = {x1, x2};
  float* relcur = relA;
  float* relnext = relB;

  for (int l = 0; l < 3; ++l) {
    float* xout = xbuf[l & 1];
    (void)hipMemsetAsync(aggin, 0, 2 * (size_t)N_NODES * H * sizeof(float), stream);
    edge_agg_kernel<<<N_EDGES / 8, 256, 0, stream>>>(xin, relcur, ei, et, norm,
                                                     aggin, aggout);
    fused_layer_kernel<<<(N_NODES / 16 + 7) / 8, 256, 0, stream>>>(
        aggin, aggout, xin, w_in[l], w_out[l], w_loop[l], loop_rel[l], bias[l],
        xout, l < 2 ? 1 : 0);
    if (l < 2) {
      rel_gemm_kernel<<<13, 32, 0, stream>>>(relcur, w_rel[l], relnext);
      set_loop_row<<<1, 128, 0, stream>>>(loop_rel[l + 1], relnext);
      float* tmp = relcur; relcur = relnext; relnext = tmp;
    }
    xin = xout;
  }

  (void)hipMemsetAsync(pool, 0, ((size_t)N_GRAPHS * H + N_GRAPHS) * sizeof(float), stream);
  pool_kernel<<<N_NODES / 8, 256, 0, stream>>>(xin, batch, pool, cnt);
  head_kernel<<<1, 256, 0, stream>>>(pool, cnt, rel_emb_table, rel_labels,
                                     lin_w, lin_b, out);
}